// RoiPoolingConv_52252572123507
// MI455X (gfx1250) — compile-verified
//
#include <hip/hip_runtime.h>

// RoiPoolingConv for gfx1250 (MI455X).
// 0-FLOP gather + NHWC->NCHW transpose. HBM floor ~575 MB (~25us @ 23.3 TB/s)
// if the 164 MB image stays L2-resident (192 MB L2) — so output stores are
// tagged non-temporal. Data moves exclusively via CDNA5 async global<->LDS DMA
// (ASYNCcnt), double-buffered so next chunk's gather overlaps this chunk's
// transposed store.

#define POOL    7
#define IMG_W   200
#define IMG_C   1024
#define NS      49              // POOL*POOL spatial samples per ROI
#define CHUNK   256             // channels staged per LDS tile
#define NCHUNK  (IMG_C / CHUNK) // 4
#define SPAD    260             // CHUNK + 4 dword pad: 16B-aligned rows, bank stride 4
#define TILE_DW (NS * SPAD)     // dwords per tile (~51 KB)
#define OFF_AREA 256            // bytes reserved for s_off[] at smem base
#define SMEM_BYTES (OFF_AREA + 2 * TILE_DW * 4)   // 102,176 B (< 320 KB/WGP, 3 WG/WGP)

__global__ __launch_bounds__(256) void roi_pool_async_kernel(
    const float* __restrict__ img,
    const float* __restrict__ rois,
    float* __restrict__ out,
    int nrois)
{
    extern __shared__ char smem[];
    int* s_off = (int*)smem;

    const int n = blockIdx.x;
    if (n >= nrois) return;
    const int t = (int)threadIdx.x;

    // ---- per-ROI sample offsets (uniform per block) ----
    if (t < NS) {
        const float* r = rois + (long long)n * 5;
        int x_min = __float2int_rn(r[1]);
        int y_min = __float2int_rn(r[2]);
        int x_max = __float2int_rn(r[3]);
        int y_max = __float2int_rn(r[4]);
        int h = y_max - y_min + 1;
        int w = x_max - x_min + 1;
        int jy = t / POOL;
        int jx = t % POOL;
        int yi = y_min + (jy * h) / POOL;
        int xi = x_min + (jx * w) / POOL;
        s_off[t] = yi * IMG_W + xi;      // feature-row index; *4096 B = global offset
    }
    __syncthreads();

    // Low 32 bits of a generic LDS pointer == LDS byte offset (aperture rule).
    const unsigned smem_base = (unsigned)(uintptr_t)(void*)smem;
    const unsigned tile0 = smem_base + OFF_AREA;
    const unsigned tile1 = tile0 + (unsigned)(TILE_DW * 4);

    // ---- Phase A issuer: async-gather 49 x CHUNK floats into tile (lds[s][c]).
    // Per-thread: fixed vec4 column k4 = t&63, sample s steps by 4 (12-13 issues).
    auto issue_loads = [&](int cb, unsigned tb) {
        const int k4   = t & 63;
        const int gcol = (cb + 4 * k4) * 4;                 // byte offset within row
        unsigned  loff = tb + (unsigned)(((t >> 6) * SPAD + 4 * k4) * 4);
        for (int s = t >> 6; s < NS; s += 4) {
            int goff = (s_off[s] << 12) + gcol;             // row*4KB + col, <2^31
            asm volatile("global_load_async_to_lds_b128 %0, %1, %2"
                         :: "v"(loff), "v"(goff), "s"(img) : "memory");
            loff += (unsigned)(4 * SPAD * 4);
        }
    };

    // ---- Phase B issuer: async transposed store tile -> out (contiguous global).
    // Element e = t + 256*i; cc = e/49, s = e%49; lds dword = s*SPAD + cc.
    // Incremental: s += 11 (mod 49), lds += 11*SPAD+5, carry fix -(49*SPAD-1).
    auto issue_stores = [&](int cb, unsigned tb) {
        int cc0 = t / NS;
        int s   = t - cc0 * NS;
        unsigned loff = tb + (unsigned)((s * SPAD + cc0) * 4);
        int goff = ((n * IMG_C + cb) * NS + t) * 4;         // bytes, <2^31, coalesced
        for (int i = 0; i < NS; ++i) {                      // 256*49 = CHUNK*NS exact
            asm volatile("global_store_async_from_lds_b32 %0, %1, %2 th:TH_STORE_NT"
                         :: "v"(goff), "v"(loff), "s"(out) : "memory");
            goff += 256 * 4;
            s    += 11;
            loff += (unsigned)((11 * SPAD + 5) * 4);
            int carry = (s >= NS);
            s    = carry ? s - NS : s;
            loff = carry ? loff - (unsigned)((NS * SPAD - 1) * 4) : loff;
        }
    };

    // ---- software pipeline: loads(i+1) overlap stores(i) ----
    issue_loads(0, tile0);
    asm volatile("s_wait_asynccnt 0x0" ::: "memory");
    __syncthreads();

    unsigned cur = tile0, nxt = tile1;
    for (int ci = 0; ci < NCHUNK; ++ci) {
        if (ci + 1 < NCHUNK) issue_loads((ci + 1) * CHUNK, nxt);
        issue_stores(ci * CHUNK, cur);
        asm volatile("s_wait_asynccnt 0x0" ::: "memory");   // drains loads+stores
        __syncthreads();                                     // tile swap safe
        unsigned tmp = cur; cur = nxt; nxt = tmp;
    }
}

extern "C" void kernel_launch(void* const* d_in, const int* in_sizes, int n_in,
                              void* d_out, int out_size, void* d_ws, size_t ws_size,
                              hipStream_t stream) {
    const float* img  = (const float*)d_in[0];   // (1, 200, 200, 1024) f32, NHWC
    const float* rois = (const float*)d_in[1];   // (N, 5) f32
    float* out = (float*)d_out;                  // (N, 1024, 7, 7) f32
    int nrois = in_sizes[1] / 5;                 // 2048

    roi_pool_async_kernel<<<nrois, 256, SMEM_BYTES, stream>>>(img, rois, out, nrois);
}